// TokenMemoryBank_8186207666893
// MI455X (gfx1250) — compile-verified
//
#include <hip/hip_runtime.h>
#include <stdint.h>

#define NGRAM  4
#define DSTATE 16
#define FNV_OFFSET 2166136261u
#define FNV_PRIME  16777619u
#define BLK 256

// CDNA5 async global->LDS staging (ASYNCcnt-tracked), guarded so the file
// compiles even if the toolchain lacks the builtins.
#if defined(__has_builtin)
#if __has_builtin(__builtin_amdgcn_global_load_async_to_lds_b128) && \
    __has_builtin(__builtin_amdgcn_s_wait_asynccnt)
#define USE_ASYNC_LDS 1
#endif
#endif
#ifndef USE_ASYNC_LDS
#define USE_ASYNC_LDS 0
#endif

#if USE_ASYNC_LDS
// Probe-derived prototype: param0 is 'int __vector(4) __device__ *', i.e. a
// non-const v4i pointer in the global address space (AS1). LDS side is AS3.
typedef int v4i __attribute__((ext_vector_type(4)));
typedef __attribute__((address_space(1))) v4i* gv4i_p;
typedef __attribute__((address_space(3))) v4i* lv4i_p;
#endif

// ---------------------------------------------------------------------------
// Kernel A: one thread per window. Hash the 4-token window (b128 load),
// record the slot, zero the slot's accumulators (count + 16 sums, b128
// stores; duplicate zero-writes across colliding windows are benign), and
// prefetch the bank row + count word into L2 for the gather pass
// (global_prefetch_b8; touched footprint ~9MB << 192MB L2).
// ---------------------------------------------------------------------------
__global__ void tmb_hash_zero(const int* __restrict__ tok,
                              int M, unsigned n_slots,
                              int*    __restrict__ addrs,
                              float*  __restrict__ hitcnt,
                              float4* __restrict__ sums4,
                              const float* __restrict__ bank,
                              const int*   __restrict__ counts) {
  int i = blockIdx.x * blockDim.x + threadIdx.x;
  if (i >= M) return;

  int4 t = ((const int4*)tok)[i];              // global_load_b128
  unsigned h = FNV_OFFSET;
  h = (h ^ (unsigned)t.x) * FNV_PRIME;
  h = (h ^ (unsigned)t.y) * FNV_PRIME;
  h = (h ^ (unsigned)t.z) * FNV_PRIME;
  h = (h ^ (unsigned)t.w) * FNV_PRIME;
  unsigned slot = h % n_slots;

  addrs[i] = (int)slot;
  hitcnt[slot] = 0.0f;
  const float4 z = make_float4(0.f, 0.f, 0.f, 0.f);
  size_t b = (size_t)slot * (DSTATE / 4);
  sums4[b + 0] = z;                            // 4x global_store_b128
  sums4[b + 1] = z;
  sums4[b + 2] = z;
  sums4[b + 3] = z;

  __builtin_prefetch(bank + (size_t)slot * DSTATE, 0, 1);   // global_prefetch_b8
  __builtin_prefetch(counts + slot, 0, 1);
}

// ---------------------------------------------------------------------------
// Kernel B: one thread per (window, float4-chunk). The dense states stream is
// staged through LDS by the async engine (GLOBAL_LOAD_ASYNC_TO_LDS_B128,
// ASYNCcnt) while the wave's LOADcnt is spent on the slot-address load; the
// wave then drains s_wait_asynccnt and fires 4 non-returning
// global_atomic_add_f32 per thread into the slot accumulators.
// ---------------------------------------------------------------------------
__global__ void tmb_scatter(const int*   __restrict__ addrs,
                            const float4* __restrict__ states4,
                            int total4,                     // M * DSTATE/4
                            float* __restrict__ hitcnt,
                            float* __restrict__ sums) {
  int tid = blockIdx.x * blockDim.x + threadIdx.x;

#if USE_ASYNC_LDS
  __shared__ float4 stage[BLK];
  if (tid < total4) {
    // async: global b128 -> LDS, overlapped with the addrs load below
    __builtin_amdgcn_global_load_async_to_lds_b128(
        (gv4i_p)(v4i*)(states4 + tid),
        (lv4i_p)(v4i*)&stage[threadIdx.x], 0, 0);
  }
#endif

  if (tid >= total4) return;
  int i = tid >> 2;        // window
  int c = tid & 3;         // float4 chunk within the 16-wide row
  int slot = addrs[i];     // LOADcnt stream, overlaps the async copy

#if USE_ASYNC_LDS
  __builtin_amdgcn_s_wait_asynccnt(0);
  float4 s = stage[threadIdx.x];               // ds_load_b128
#else
  float4 s = states4[tid];                     // global_load_b128 fallback
#endif

  float* dst = sums + (size_t)slot * DSTATE + (size_t)c * 4;
  atomicAdd(dst + 0, s.x);                     // no-return global_atomic_add_f32
  atomicAdd(dst + 1, s.y);
  atomicAdd(dst + 2, s.z);
  atomicAdd(dst + 3, s.w);
  if (c == 0) atomicAdd(&hitcnt[slot], 1.0f);
}

// ---------------------------------------------------------------------------
// Kernel C: one thread per (window, float4-chunk). 4 consecutive lanes cover
// one 64B slot row -> coalesced b128 gathers of sums and bank (both L2-hot).
//   out = alpha*bank + (1-alpha)*(sum/cnt),  alpha = counts==0 ? 0 : 0.9
// cnt >= 1 is guaranteed at gathered slots (this window contributed).
// ---------------------------------------------------------------------------
__global__ void tmb_gather(const int*   __restrict__ addrs,
                           const float* __restrict__ bank,
                           const int*   __restrict__ counts,
                           const float* __restrict__ hitcnt,
                           const float* __restrict__ sums,
                           int total4,                      // M * DSTATE/4
                           float4* __restrict__ out4) {
  int tid = blockIdx.x * blockDim.x + threadIdx.x;
  if (tid >= total4) return;
  int i = tid >> 2;
  int c = tid & 3;
  int slot = addrs[i];

  float cnt   = hitcnt[slot];
  float inv   = 1.0f / cnt;
  float alpha = (counts[slot] == 0) ? 0.0f : 0.9f;
  float beta  = (1.0f - alpha) * inv;

  size_t off = (size_t)slot * DSTATE + (size_t)c * 4;
  float4 s4 = *(const float4*)(sums + off);   // global_load_b128
  float4 b4 = *(const float4*)(bank + off);   // global_load_b128 (prefetched)

  float4 o;
  o.x = fmaf(alpha, b4.x, beta * s4.x);
  o.y = fmaf(alpha, b4.y, beta * s4.y);
  o.z = fmaf(alpha, b4.z, beta * s4.z);
  o.w = fmaf(alpha, b4.w, beta * s4.w);
  out4[tid] = o;                               // global_store_b128
}

// ---------------------------------------------------------------------------
// Launch. Inputs (setup_inputs order):
//   d_in[0] token_window (16,8192,4) int32
//   d_in[1] states       (16,8192,16) float32
//   d_in[2] bank         (N_SLOTS,16) float32
//   d_in[3] counts       (N_SLOTS,)   int32
// Output: read_out (16,8192,16) float32.
// Workspace: [sums: n_slots*16*f32][hitcnt: n_slots*f32][addrs: M*i32]
// ---------------------------------------------------------------------------
extern "C" void kernel_launch(void* const* d_in, const int* in_sizes, int n_in,
                              void* d_out, int out_size, void* d_ws, size_t ws_size,
                              hipStream_t stream) {
  const int*   tok    = (const int*)  d_in[0];
  const float* states = (const float*)d_in[1];
  const float* bank   = (const float*)d_in[2];
  const int*   counts = (const int*)  d_in[3];

  const int      M       = in_sizes[0] / NGRAM;     // 131072 windows
  const unsigned n_slots = (unsigned)in_sizes[3];   // 2,000,000

  char* ws = (char*)d_ws;
  size_t off = 0;
  float* sums = (float*)(ws + off);
  off += (size_t)n_slots * DSTATE * sizeof(float);
  off = (off + 255) & ~(size_t)255;
  float* hitcnt = (float*)(ws + off);
  off += (size_t)n_slots * sizeof(float);
  off = (off + 255) & ~(size_t)255;
  int* addrs = (int*)(ws + off);

  int gridA = (M + BLK - 1) / BLK;
  tmb_hash_zero<<<gridA, BLK, 0, stream>>>(tok, M, n_slots, addrs, hitcnt,
                                           (float4*)sums, bank, counts);

  int total4 = M * (DSTATE / 4);
  int gridBC = (total4 + BLK - 1) / BLK;
  tmb_scatter<<<gridBC, BLK, 0, stream>>>(addrs, (const float4*)states, total4,
                                          hitcnt, sums);

  tmb_gather<<<gridBC, BLK, 0, stream>>>(addrs, bank, counts, hitcnt, sums,
                                         total4, (float4*)d_out);
}